// MetapathLayer_8048768713043
// MI455X (gfx1250) — compile-verified
//
#include <hip/hip_runtime.h>
#include <hip/hip_bf16.h>

typedef __attribute__((ext_vector_type(2))) float v2f;
typedef __attribute__((ext_vector_type(8))) float v8f;

#define N_NODES 50000
#define DIM     128
#define NEDGES  800000
#define LN_EPS  1e-5f

// ---------------------------------------------------------------------------
// Kernel 0: zero workspace region (S + cnt), float4 stores
// ---------------------------------------------------------------------------
__global__ __launch_bounds__(256) void zero_kernel(float4* __restrict__ p, int n4) {
    int i = blockIdx.x * blockDim.x + threadIdx.x;
    if (i < n4) p[i] = make_float4(0.f, 0.f, 0.f, 0.f);
}

// ---------------------------------------------------------------------------
// Kernel 1: fold the two linear layers:  A = out_w @ wv,  b_tot = out_w @ bv + out_b
// wv = in_proj_w rows [2D, 3D), bv = in_proj_b[2D:3D)
// grid = 128 blocks (output row i), block = 128 threads (output col j)
// ---------------------------------------------------------------------------
__global__ __launch_bounds__(128) void fold_weights_kernel(
    const float* __restrict__ in_w, const float* __restrict__ in_b,
    const float* __restrict__ out_w, const float* __restrict__ out_b,
    float* __restrict__ Amat, float* __restrict__ btot)
{
    const int i = blockIdx.x;
    const int j = threadIdx.x;
    const float* ow = out_w + (size_t)i * DIM;
    float acc = 0.f;
    #pragma unroll 4
    for (int d = 0; d < DIM; ++d)
        acc = fmaf(ow[d], in_w[(size_t)(2 * DIM + d) * DIM + j], acc);
    Amat[(size_t)i * DIM + j] = acc;
    if (j == 0) {
        float b = out_b[i];
        for (int d = 0; d < DIM; ++d)
            b = fmaf(ow[d], in_b[2 * DIM + d], b);
        btot[i] = b;
    }
}

// ---------------------------------------------------------------------------
// Kernel 2: edge scatter. One wave32 per edge: each lane moves float4 (128 f32
// per row), atomic f32 adds into S[dst]; lane 0 bumps cnt[dst].
// x_src (25.6 MB) and S (25.6 MB) are L2-resident on MI455X (192 MB L2).
// ---------------------------------------------------------------------------
__global__ __launch_bounds__(256) void scatter_kernel(
    const float* __restrict__ x_src, const int* __restrict__ eidx,
    float* __restrict__ S, float* __restrict__ cnt)
{
    const int gw   = blockIdx.x * 8 + (threadIdx.x >> 5);
    const int lane = threadIdx.x & 31;
    if (gw >= NEDGES) return;
    const int src = eidx[gw];
    const int dst = eidx[NEDGES + gw];
    const float4 v = *(const float4*)(x_src + (size_t)src * DIM + lane * 4);
    float* d = S + (size_t)dst * DIM + lane * 4;
    atomicAdd(d + 0, v.x);
    atomicAdd(d + 1, v.y);
    atomicAdd(d + 2, v.z);
    atomicAdd(d + 3, v.w);
    if (lane == 0) atomicAdd(cnt + dst, 1.0f);
}

// ---------------------------------------------------------------------------
// Kernel 3: fused  (S/max(cnt,1)) @ A^T + b_tot*(cnt/max(cnt,1)) + x_dst, then
// LayerNorm.  256 threads = 8 waves; block tile = 16 nodes x 128 cols.
// Each wave: one 16x16 tile via 32x V_WMMA_F32_16X16X4_F32 (K=128).
//
// f32 WMMA fragment layouts (ISA 7.12.2):
//   A 16x4:  lanes 0-15 M=0..15 {K=k,k+1}; lanes 16-31 M=0..15 {K=k+2,k+3}
//   B 4x16:  lanes 0-15 N=0..15 {K=k,k+1}; lanes 16-31 N=0..15 {K=k+2,k+3}
//   C/D 16x16: VGPR r, lanes 0-15 -> M=r, lanes 16-31 -> M=r+8, N=lane%16
// ---------------------------------------------------------------------------
__global__ __launch_bounds__(256) void gemm_ln_kernel(
    const float* __restrict__ S,    const float* __restrict__ cnt,
    const float* __restrict__ Amat, const float* __restrict__ btot,
    const float* __restrict__ x_dst,
    const float* __restrict__ ln_g, const float* __restrict__ ln_b,
    float* __restrict__ out)
{
    __shared__ float As[16][132];   // scaled S tile, padded stride (bank spread)
    __shared__ float Hs[16][128];   // pre-LN tile
    __shared__ float sc[16];        // 1/max(cnt,1)
    __shared__ float bsc[16];       // cnt/max(cnt,1)

    const int tid  = threadIdx.x;
    const int w    = tid >> 5;
    const int lane = tid & 31;
    const int half = lane >> 4;
    const int l16  = lane & 15;
    const int row0 = blockIdx.x * 16;     // N = 50000 = 16 * 3125, exact tiling

    if (tid < 16) {
        const float c  = cnt[row0 + tid];
        const float cm = fmaxf(c, 1.0f);
        sc[tid]  = 1.0f / cm;
        bsc[tid] = c / cm;
    }
    __syncthreads();

    // Stage scaled S tile: 16 rows x 128 cols; 32 lanes x float4 per row.
    #pragma unroll
    for (int p = 0; p < 2; ++p) {
        const int r  = (tid >> 5) + p * 8;
        const int c4 = (tid & 31) * 4;
        const float4 v = *(const float4*)(S + (size_t)(row0 + r) * DIM + c4);
        const float s = sc[r];
        As[r][c4 + 0] = v.x * s;
        As[r][c4 + 1] = v.y * s;
        As[r][c4 + 2] = v.z * s;
        As[r][c4 + 3] = v.w * s;
    }
    __syncthreads();

    // GEMM: wave w owns output columns [w*16, w*16+16)
    const int n0 = w * 16;
    const int n  = n0 + l16;
    const float* __restrict__ Brow = Amat + (size_t)n * DIM;  // B[k][n] = A[n][k]
    v8f acc = {};
    #pragma unroll
    for (int k = 0; k < DIM; k += 4) {
        const int kk = k + 2 * half;
        v2f a, b;
        a.x = As[l16][kk];
        a.y = As[l16][kk + 1];
        b.x = Brow[kk];
        b.y = Brow[kk + 1];
        acc = __builtin_amdgcn_wmma_f32_16x16x4_f32(
            /*neg_a=*/false, a, /*neg_b=*/false, b,
            /*c_mod=*/(short)0, acc, /*reuse_a=*/false, /*reuse_b=*/false);
    }

    // Epilogue: bias * (cnt/max(cnt,1)) + residual, park in LDS for LayerNorm
    const float bt = btot[n];
    #pragma unroll
    for (int r = 0; r < 8; ++r) {
        const int m  = r + half * 8;
        const int gm = row0 + m;
        Hs[m][n] = acc[r] + bt * bsc[m] + x_dst[(size_t)gm * DIM + n];
    }
    __syncthreads();

    // LayerNorm: wave w handles rows 2w, 2w+1; 4 elems/lane, wave32 shfl reduce
    #pragma unroll
    for (int rr = 0; rr < 2; ++rr) {
        const int m  = w * 2 + rr;
        const int c0 = lane * 4;
        const float4 h = *(const float4*)&Hs[m][c0];
        float sum = h.x + h.y + h.z + h.w;
        float sq  = h.x * h.x + h.y * h.y + h.z * h.z + h.w * h.w;
        #pragma unroll
        for (int off = 16; off >= 1; off >>= 1) {
            sum += __shfl_xor(sum, off, 32);
            sq  += __shfl_xor(sq,  off, 32);
        }
        const float mean = sum * (1.0f / 128.0f);
        const float var  = sq * (1.0f / 128.0f) - mean * mean;
        const float rstd = rsqrtf(var + LN_EPS);
        float4 o;
        o.x = (h.x - mean) * rstd * ln_g[c0 + 0] + ln_b[c0 + 0];
        o.y = (h.y - mean) * rstd * ln_g[c0 + 1] + ln_b[c0 + 1];
        o.z = (h.z - mean) * rstd * ln_g[c0 + 2] + ln_b[c0 + 2];
        o.w = (h.w - mean) * rstd * ln_g[c0 + 3] + ln_b[c0 + 3];
        *(float4*)(out + (size_t)(row0 + m) * DIM + c0) = o;
    }
}

// ---------------------------------------------------------------------------
extern "C" void kernel_launch(void* const* d_in, const int* in_sizes, int n_in,
                              void* d_out, int out_size, void* d_ws, size_t ws_size,
                              hipStream_t stream) {
    const float* x_src  = (const float*)d_in[0];
    const float* x_dst  = (const float*)d_in[1];
    const int*   eidx   = (const int*)  d_in[2];
    const float* in_w   = (const float*)d_in[3];
    const float* in_b   = (const float*)d_in[4];
    const float* out_w  = (const float*)d_in[5];
    const float* out_b  = (const float*)d_in[6];
    const float* ln_g   = (const float*)d_in[7];
    const float* ln_b   = (const float*)d_in[8];
    float* out = (float*)d_out;

    // Workspace layout (floats): S[N*D] | cnt[N] | Amat[D*D] | btot[D]
    float* S    = (float*)d_ws;
    float* cnt  = S + (size_t)N_NODES * DIM;
    float* Amat = cnt + N_NODES;
    float* btot = Amat + DIM * DIM;

    // 0) zero S + cnt (contiguous)
    const int n4 = (N_NODES * DIM + N_NODES) / 4;
    zero_kernel<<<(n4 + 255) / 256, 256, 0, stream>>>((float4*)S, n4);

    // 1) fold weights: A = out_w @ wv, b_tot = out_w @ bv + out_b
    fold_weights_kernel<<<DIM, DIM, 0, stream>>>(in_w, in_b, out_w, out_b, Amat, btot);

    // 2) edge scatter-add (one wave per edge)
    scatter_kernel<<<(NEDGES + 7) / 8, 256, 0, stream>>>(x_src, eidx, S, cnt);

    // 3) fused WMMA GEMM + bias + residual + LayerNorm
    gemm_ln_kernel<<<N_NODES / 16, 256, 0, stream>>>(S, cnt, Amat, btot,
                                                     x_dst, ln_g, ln_b, out);
}